// GATConv_66202625900915
// MI455X (gfx1250) — compile-verified
//
#include <hip/hip_runtime.h>
#include <hip/hip_bf16.h>

// ---------- constants (match reference) ----------
#define IN_F 128
#define HD   128   // H*D
#define NH   4     // heads
#define DH   32    // dims per head
#define NEG_SLOPE 0.2f

typedef __attribute__((ext_vector_type(2))) float v2f;
typedef __attribute__((ext_vector_type(8))) float v8f;

// ---------- helpers ----------
__device__ __forceinline__ unsigned f2ord(float f) {
    unsigned b = __float_as_uint(f);
    return (b & 0x80000000u) ? ~b : (b | 0x80000000u);
}
__device__ __forceinline__ float ord2f(unsigned k) {
    unsigned b = (k & 0x80000000u) ? (k ^ 0x80000000u) : ~k;
    return __uint_as_float(b);
}

// ---------- 0: fill ----------
__global__ void fill_u32(unsigned* __restrict__ p, unsigned v, long long n) {
    long long t = (long long)blockIdx.x * blockDim.x + threadIdx.x;
    if (t < n) p[t] = v;
}

// ---------- 1: ft = feat @ W via V_WMMA_F32_16X16X4_F32 ----------
// One wave computes one 16x16 tile of ft. Block = 256 threads = 8 waves
// covering the 8 column tiles (128 cols) of one 16-row strip.
__global__ __launch_bounds__(256) void gemm_wmma_f32(
    const float* __restrict__ A,   // [N, 128] feat
    const float* __restrict__ B,   // [128, 128] W
    float* __restrict__ C,         // [N, 128] ft
    int nrows)
{
    const int lane    = threadIdx.x & 31;
    const int wave    = threadIdx.x >> 5;
    const int l15     = lane & 15;
    const int halfSel = lane >> 4;          // 0: K pair {0,1}, 1: K pair {2,3}
    const int mBase   = blockIdx.x * 16;
    const int nBase   = wave * 16;

    int row = mBase + l15;
    if (row >= nrows) row = nrows - 1;      // clamp loads; stores are guarded
    const int col = nBase + l15;

    const float* aptr = A + (size_t)row * IN_F + (halfSel << 1);
    const float* bptr = B + (size_t)(halfSel << 1) * HD + col;

    v8f acc = {};
#pragma unroll
    for (int k = 0; k < IN_F; k += 4) {
        v2f a;                               // A[row][k + 2*halfSel + {0,1}]
        a.x = aptr[k];
        a.y = aptr[k + 1];
        v2f b;                               // B[k + 2*halfSel + {0,1}][col]
        b.x = bptr[(size_t)k * HD];
        b.y = bptr[(size_t)(k + 1) * HD];
        acc = __builtin_amdgcn_wmma_f32_16x16x4_f32(
            /*neg_a=*/false, a, /*neg_b=*/false, b,
            /*c_mod=*/(short)0, acc, /*reuse_a=*/false, /*reuse_b=*/false);
    }

    // C/D layout: VGPR r -> lanes 0-15: M=r, lanes 16-31: M=r+8; N = lane&15
#pragma unroll
    for (int r = 0; r < 8; ++r) {
        const int m = mBase + r + (halfSel << 3);
        if (m < nrows) C[(size_t)m * HD + nBase + l15] = acc[r];
    }
}

// ---------- 2: el/er per (node, head) ----------
__global__ void node_attn(const float* __restrict__ ft,
                          const float* __restrict__ attn_l,
                          const float* __restrict__ attn_r,
                          float* __restrict__ el, float* __restrict__ er,
                          long long nh_total)
{
    long long t = (long long)blockIdx.x * blockDim.x + threadIdx.x;
    if (t >= nh_total) return;
    const long long n = t >> 2;
    const int h = (int)(t & 3);
    const float* f  = ft + (size_t)n * HD + h * DH;
    const float* al = attn_l + h * DH;
    const float* ar = attn_r + h * DH;
    float sl = 0.f, sr = 0.f;
#pragma unroll
    for (int d = 0; d < DH; d += 4) {
        const float4 fv = *(const float4*)(f + d);
        const float4 lv = *(const float4*)(al + d);
        const float4 rv = *(const float4*)(ar + d);
        sl += fv.x * lv.x + fv.y * lv.y + fv.z * lv.z + fv.w * lv.w;
        sr += fv.x * rv.x + fv.y * rv.y + fv.z * rv.z + fv.w * rv.w;
    }
    el[t] = sl;
    er[t] = sr;
}

// ---------- 3: edge logits + LeakyReLU + segment max ----------
__global__ void edge_logits(const float* __restrict__ el,
                            const float* __restrict__ er,
                            const int* __restrict__ src,
                            const int* __restrict__ dst,
                            float* __restrict__ ebuf,
                            unsigned* __restrict__ emaxk,
                            long long total)           // E*4
{
    long long t = (long long)blockIdx.x * blockDim.x + threadIdx.x;
    if (t >= total) return;
    const long long e = t >> 2;
    const int h = (int)(t & 3);
    const int s = src[e], d = dst[e];
    float v = el[(size_t)s * NH + h] + er[(size_t)d * NH + h];
    v = v > 0.f ? v : NEG_SLOPE * v;
    ebuf[t] = v;
    atomicMax(&emaxk[(size_t)d * NH + h], f2ord(v));
}

// ---------- 4: decode ordered-uint max; guard isolated nodes ----------
__global__ void finalize_max(const unsigned* __restrict__ emaxk,
                             float* __restrict__ emaxf, long long nh_total)
{
    long long t = (long long)blockIdx.x * blockDim.x + threadIdx.x;
    if (t >= nh_total) return;
    const unsigned k = emaxk[t];
    float v = (k == 0u) ? 0.f : ord2f(k);
    if (!__builtin_isfinite(v)) v = 0.f;
    emaxf[t] = v;
}

// ---------- 5: exponentials + segment sum ----------
__global__ void edge_exp(const float* __restrict__ emaxf,
                         const int* __restrict__ dst,
                         float* __restrict__ ebuf,
                         float* __restrict__ denom,
                         long long total)              // E*4
{
    long long t = (long long)blockIdx.x * blockDim.x + threadIdx.x;
    if (t >= total) return;
    const long long e = t >> 2;
    const int h = (int)(t & 3);
    const int d = dst[e];
    const float ex = expf(ebuf[t] - emaxf[(size_t)d * NH + h]);
    ebuf[t] = ex;
    atomicAdd(&denom[(size_t)d * NH + h], ex);
}

// ---------- 6: guarded reciprocal denominators ----------
__global__ void finalize_denom(float* __restrict__ denom, long long nh_total) {
    long long t = (long long)blockIdx.x * blockDim.x + threadIdx.x;
    if (t >= nh_total) return;
    const float s = denom[t];
    denom[t] = (s > 0.f) ? 1.0f / s : 1.0f;
}

// ---------- 7: weighted gather-scatter aggregation ----------
// One thread per (edge, 4-elem chunk): 32 chunks cover H*D = 128 elements.
__global__ void edge_aggregate(const float* __restrict__ ft,
                               const float* __restrict__ ebuf,
                               const float* __restrict__ rdenom,
                               const int* __restrict__ src,
                               const int* __restrict__ dst,
                               float* __restrict__ out,
                               long long total)        // E*32
{
    long long t = (long long)blockIdx.x * blockDim.x + threadIdx.x;
    if (t >= total) return;
    const long long e = t >> 5;
    const int q = (int)(t & 31);            // chunk index; head = q>>3
    const int h = q >> 3;
    const int s = src[e], d = dst[e];
    const float a = ebuf[e * NH + h] * rdenom[(size_t)d * NH + h];
    const float4 v = ((const float4*)(ft + (size_t)s * HD))[q];
    float* o = out + (size_t)d * HD + q * 4;
    atomicAdd(o + 0, a * v.x);
    atomicAdd(o + 1, a * v.y);
    atomicAdd(o + 2, a * v.z);
    atomicAdd(o + 3, a * v.w);
}

// ---------- host launcher ----------
static inline unsigned nblk(long long n, int bs) {
    return (unsigned)((n + bs - 1) / bs);
}

extern "C" void kernel_launch(void* const* d_in, const int* in_sizes, int n_in,
                              void* d_out, int out_size, void* d_ws, size_t ws_size,
                              hipStream_t stream) {
    const float* feat   = (const float*)d_in[0];
    const float* W      = (const float*)d_in[1];
    const float* attn_l = (const float*)d_in[2];
    const float* attn_r = (const float*)d_in[3];
    const int*   src    = (const int*)d_in[4];
    const int*   dst    = (const int*)d_in[5];
    float*       out    = (float*)d_out;

    const long long N = in_sizes[0] / IN_F;
    const long long E = in_sizes[4];

    // workspace layout
    float*    ft    = (float*)d_ws;                         // N*128
    float*    el    = ft + (size_t)N * HD;                  // N*4
    float*    er    = el + (size_t)N * NH;                  // N*4
    unsigned* emaxk = (unsigned*)(er + (size_t)N * NH);     // N*4
    float*    emaxf = (float*)(emaxk + (size_t)N * NH);     // N*4
    float*    denom = emaxf + (size_t)N * NH;               // N*4
    float*    ebuf  = denom + (size_t)N * NH;               // E*4

    const int BS = 256;
    const long long NHtot = N * NH;
    const long long EHtot = E * NH;
    const long long EAgg  = E * 32;

    // 0: init accumulators
    fill_u32<<<nblk((long long)out_size, BS), BS, 0, stream>>>((unsigned*)out, 0u, out_size);
    fill_u32<<<nblk(NHtot, BS), BS, 0, stream>>>((unsigned*)denom, 0u, NHtot);
    fill_u32<<<nblk(NHtot, BS), BS, 0, stream>>>(emaxk, 0u, NHtot);

    // 1: projection GEMM on the WMMA pipe
    gemm_wmma_f32<<<nblk(N, 16), 256, 0, stream>>>(feat, W, ft, (int)N);

    // 2: per-node attention dots
    node_attn<<<nblk(NHtot, BS), BS, 0, stream>>>(ft, attn_l, attn_r, el, er, NHtot);

    // 3: edge logits + segment max
    edge_logits<<<nblk(EHtot, BS), BS, 0, stream>>>(el, er, src, dst, ebuf, emaxk, EHtot);

    // 4: decode max
    finalize_max<<<nblk(NHtot, BS), BS, 0, stream>>>(emaxk, emaxf, NHtot);

    // 5: exp + segment sum
    edge_exp<<<nblk(EHtot, BS), BS, 0, stream>>>(emaxf, dst, ebuf, denom, EHtot);

    // 6: reciprocal denominators
    finalize_denom<<<nblk(NHtot, BS), BS, 0, stream>>>(denom, NHtot);

    // 7: weighted aggregation (memory/atomic bound stage)
    edge_aggregate<<<nblk(EAgg, BS), BS, 0, stream>>>(ft, ebuf, denom, src, dst, out, EAgg);
}